// DrugEncoderGNN_35244501631134
// MI455X (gfx1250) — compile-verified
//
#include <hip/hip_runtime.h>

// CDNA5 fp32 WMMA fragments (wave32):
//   A 16x4  -> 2 VGPRs/lane, B 4x16 -> 2 VGPRs/lane, C/D 16x16 -> 8 VGPRs/lane
typedef __attribute__((ext_vector_type(2))) float v2f;
typedef __attribute__((ext_vector_type(8))) float v8f;

#define HDIM 128
#define LDS_STRIDE 132   // 128 + 4 pad: 16 rows hit 16 distinct LDS banks

// ---------------- degree / dinv ----------------
__global__ void deg_kernel(const int* __restrict__ dst, float* __restrict__ deg, int E) {
    int t = blockIdx.x * blockDim.x + threadIdx.x;
    if (t < E) atomicAdd(&deg[dst[t]], 1.0f);
}

__global__ void dinv_kernel(float* deg, int N) {
    int t = blockIdx.x * blockDim.x + threadIdx.x;
    if (t < N) deg[t] = rsqrtf(deg[t] + 1.0f);   // deg includes self-loop
}

// ---------------- zero-pad W (K x 128) into Wp (KP x 128) ----------------
__global__ void pad_w_kernel(const float* __restrict__ W, float* __restrict__ Wp,
                             int K, int total /* = KP*HDIM */) {
    int t = blockIdx.x * blockDim.x + threadIdx.x;
    if (t >= total) return;
    int r = t >> 7;                       // row (HDIM == 128)
    Wp[t] = (r < K) ? W[t] : 0.0f;
}

// ---------------- WMMA GEMM: h = X@W ; agg = h * dinv^2 (self-loop seed) ----
// One 16-row block per 256-thread workgroup; wave w computes cols [16w,16w+16).
// KX = true column count of X (row stride); KP = padded K (multiple of 8).
// W must have KP rows (zero-padded), so the WMMA loop is branch-free and EXEC
// stays all-1s (ISA requirement). X and agg may alias (in-place layer update):
// X is fully staged to LDS before any global store, and each block reads only
// the rows it writes.
template <int KX, int KP>
__global__ void gcn_gemm_wmma(const float* X, const float* __restrict__ W,
                              const float* __restrict__ dinv,
                              float* __restrict__ h, float* agg) {
    __shared__ float xt[16 * LDS_STRIDE];
    const int tid  = threadIdx.x;
    const int row0 = blockIdx.x * 16;

    // stage 16 x KP tile of X (zero-pad cols KX..KP)
    for (int idx = tid; idx < 16 * KP; idx += 256) {
        const int r = idx / KP, c = idx - r * KP;   // KP is a power of two
        float v = 0.0f;
        if (c < KX) v = X[(size_t)(row0 + r) * KX + c];
        xt[r * LDS_STRIDE + c] = v;
    }
    __syncthreads();

    const int lane  = tid & 31;
    const int l15   = lane & 15;
    const int lhalf = lane >> 4;           // 0 or 1
    const int col0  = (tid >> 5) * 16;     // wave id * 16
    const float* __restrict__ Wc = W + col0 + l15;

    v8f acc0 = {}, acc1 = {};
    #pragma unroll
    for (int k = 0; k < KP; k += 8) {
        {   // chunk k..k+3 -> acc0
            const int ka = k + lhalf * 2;
            v2f a, b;
            a.x = xt[l15 * LDS_STRIDE + ka];
            a.y = xt[l15 * LDS_STRIDE + ka + 1];
            b.x = Wc[(size_t)ka * HDIM];
            b.y = Wc[(size_t)(ka + 1) * HDIM];
            acc0 = __builtin_amdgcn_wmma_f32_16x16x4_f32(
                false, a, false, b, (short)0, acc0, false, false);
        }
        {   // chunk k+4..k+7 -> acc1 (independent chain hides WMMA RAW latency)
            const int ka = k + 4 + lhalf * 2;
            v2f a, b;
            a.x = xt[l15 * LDS_STRIDE + ka];
            a.y = xt[l15 * LDS_STRIDE + ka + 1];
            b.x = Wc[(size_t)ka * HDIM];
            b.y = Wc[(size_t)(ka + 1) * HDIM];
            acc1 = __builtin_amdgcn_wmma_f32_16x16x4_f32(
                false, a, false, b, (short)0, acc1, false, false);
        }
    }

    // C/D layout: VGPR v, lanes 0-15 -> (M=v, N=lane); lanes 16-31 -> (M=v+8)
    #pragma unroll
    for (int v = 0; v < 8; ++v) {
        const int row = row0 + v + lhalf * 8;
        const size_t o = (size_t)row * HDIM + col0 + l15;
        const float val = acc0[v] + acc1[v];
        const float dv  = dinv[row];
        h[o]   = val;
        agg[o] = val * dv * dv;
    }
}

// Fallback (not expected to run; only if D_IN != 30): plain per-element GEMM.
__global__ void gcn_gemm_naive(const float* X, const float* __restrict__ W,
                               const float* __restrict__ dinv,
                               float* __restrict__ h, float* agg, int N, int K) {
    int t = blockIdx.x * blockDim.x + threadIdx.x;
    int row = t >> 7, col = t & (HDIM - 1);
    if (row >= N) return;
    float s = 0.0f;
    for (int k = 0; k < K; ++k) s += X[(size_t)row * K + k] * W[(size_t)k * HDIM + col];
    size_t o = (size_t)row * HDIM + col;
    float dv = dinv[row];
    h[o] = s;
    agg[o] = s * dv * dv;
}

// ---------------- edge scatter: agg[dst] += h[src] * dinv[src]*dinv[dst] ----
__global__ void edge_kernel(const float* __restrict__ h,
                            const float* __restrict__ dinv,
                            const int* __restrict__ src, const int* __restrict__ dst,
                            float* __restrict__ agg, int E) {
    int t = blockIdx.x * blockDim.x + threadIdx.x;
    int e = t >> 5;
    if (e >= E) return;
    int lane = t & 31;
    int s = src[e], d = dst[e];
    float norm = dinv[s] * dinv[d];
    float4 v = ((const float4*)(h + (size_t)s * HDIM))[lane];
    float* a = agg + (size_t)d * HDIM + lane * 4;
    atomicAdd(a + 0, v.x * norm);
    atomicAdd(a + 1, v.y * norm);
    atomicAdd(a + 2, v.z * norm);
    atomicAdd(a + 3, v.w * norm);
}

// ---------------- act = relu(agg + b), in place ----------------
__global__ void relu_bias_kernel(float* __restrict__ agg,
                                 const float* __restrict__ bias, long n4) {
    long t = (long)blockIdx.x * blockDim.x + threadIdx.x;
    if (t >= n4) return;
    float4 v = ((float4*)agg)[t];
    float4 b = ((const float4*)bias)[t & 31];   // 32 float4 per 128-wide row
    v.x = fmaxf(v.x + b.x, 0.0f);
    v.y = fmaxf(v.y + b.y, 0.0f);
    v.z = fmaxf(v.z + b.z, 0.0f);
    v.w = fmaxf(v.w + b.w, 0.0f);
    ((float4*)agg)[t] = v;
}

// ---------------- global mean pool ----------------
__global__ void pool_kernel(const float* __restrict__ act,
                            const int* __restrict__ batch,
                            float* __restrict__ sums, int N) {
    int t = blockIdx.x * blockDim.x + threadIdx.x;
    int node = t >> 5;
    if (node >= N) return;
    int lane = t & 31;
    int g = batch[node];
    float4 v = ((const float4*)(act + (size_t)node * HDIM))[lane];
    float* s = sums + (size_t)g * HDIM + lane * 4;
    atomicAdd(s + 0, v.x);
    atomicAdd(s + 1, v.y);
    atomicAdd(s + 2, v.z);
    atomicAdd(s + 3, v.w);
}

__global__ void count_kernel(const int* __restrict__ batch,
                             float* __restrict__ counts, int N) {
    int t = blockIdx.x * blockDim.x + threadIdx.x;
    if (t < N) atomicAdd(&counts[batch[t]], 1.0f);
}

__global__ void finalize_kernel(const float* __restrict__ sums,
                                const float* __restrict__ counts,
                                float* __restrict__ out, int total) {
    int t = blockIdx.x * blockDim.x + threadIdx.x;
    if (t >= total) return;
    float c = fmaxf(counts[t >> 7], 1.0f);      // HDIM == 128
    out[t] = sums[t] / c;
}

extern "C" void kernel_launch(void* const* d_in, const int* in_sizes, int n_in,
                              void* d_out, int out_size, void* d_ws, size_t ws_size,
                              hipStream_t stream) {
    const float* x     = (const float*)d_in[0];
    const int*   edges = (const int*)  d_in[1];
    const int*   batch = (const int*)  d_in[2];
    const float* W1 = (const float*)d_in[3];
    const float* b1 = (const float*)d_in[4];
    const float* W2 = (const float*)d_in[5];
    const float* b2 = (const float*)d_in[6];
    const float* W3 = (const float*)d_in[7];
    const float* b3 = (const float*)d_in[8];
    float* out = (float*)d_out;

    const int N = in_sizes[2];                 // 200000 (multiple of 16)
    const int E = in_sizes[1] / 2;             // 600000
    const int D = in_sizes[0] / N;             // 30
    const int G = out_size / HDIM;             // 8192

    const int* src = edges;
    const int* dst = edges + E;

    // workspace: dinv | h | agg(act) | sums | counts | W1p(32x128)
    float* ws     = (float*)d_ws;
    float* dinv   = ws;
    float* h      = dinv + N;
    float* agg    = h    + (size_t)N * HDIM;
    float* sums   = agg  + (size_t)N * HDIM;
    float* counts = sums + (size_t)G * HDIM;
    float* W1p    = counts + G;

    hipMemsetAsync(dinv,   0, sizeof(float) * (size_t)N,        stream);
    hipMemsetAsync(sums,   0, sizeof(float) * (size_t)G * HDIM, stream);
    hipMemsetAsync(counts, 0, sizeof(float) * (size_t)G,        stream);

    deg_kernel <<<(E + 255) / 256, 256, 0, stream>>>(dst, dinv, E);
    dinv_kernel<<<(N + 255) / 256, 256, 0, stream>>>(dinv, N);

    const int gemmBlocks = N / 16;
    const int edgeBlocks = (E * 32 + 255) / 256;
    const long n4        = (long)N * (HDIM / 4);
    const int reluBlocks = (int)((n4 + 255) / 256);
    const int poolBlocks = (N * 32 + 255) / 256;

    // ---- layer 1 (K = 30, weights zero-padded to 32 rows) ----
    if (D == 30) {
        pad_w_kernel<<<(32 * HDIM + 255) / 256, 256, 0, stream>>>(W1, W1p, 30, 32 * HDIM);
        gcn_gemm_wmma<30, 32><<<gemmBlocks, 256, 0, stream>>>(x, W1p, dinv, h, agg);
    } else {
        gcn_gemm_naive<<<((size_t)N * HDIM + 255) / 256, 256, 0, stream>>>(
            x, W1, dinv, h, agg, N, D);
    }
    edge_kernel     <<<edgeBlocks, 256, 0, stream>>>(h, dinv, src, dst, agg, E);
    relu_bias_kernel<<<reluBlocks, 256, 0, stream>>>(agg, b1, n4);

    // ---- layer 2 (K = 128), agg is both input activation and output (safe) ----
    gcn_gemm_wmma<HDIM, HDIM><<<gemmBlocks, 256, 0, stream>>>(agg, W2, dinv, h, agg);
    edge_kernel     <<<edgeBlocks, 256, 0, stream>>>(h, dinv, src, dst, agg, E);
    relu_bias_kernel<<<reluBlocks, 256, 0, stream>>>(agg, b2, n4);

    // ---- layer 3 ----
    gcn_gemm_wmma<HDIM, HDIM><<<gemmBlocks, 256, 0, stream>>>(agg, W3, dinv, h, agg);
    edge_kernel     <<<edgeBlocks, 256, 0, stream>>>(h, dinv, src, dst, agg, E);
    relu_bias_kernel<<<reluBlocks, 256, 0, stream>>>(agg, b3, n4);

    // ---- global mean pool ----
    pool_kernel <<<poolBlocks, 256, 0, stream>>>(agg, batch, sums, N);
    count_kernel<<<(N + 255) / 256, 256, 0, stream>>>(batch, counts, N);
    finalize_kernel<<<(out_size + 255) / 256, 256, 0, stream>>>(sums, counts, out, out_size);
}